// TernaryModule_71012989272075
// MI455X (gfx1250) — compile-verified
//
#include <hip/hip_runtime.h>
#include <stdint.h>

typedef float v2f __attribute__((ext_vector_type(2)));
typedef float v8f __attribute__((ext_vector_type(8)));
typedef int   v4i __attribute__((ext_vector_type(4)));

#define ROW        8192
#define BLOCK      256
#define WAVES      (BLOCK / 32)
#define WAVE_ELEMS (ROW / WAVES)      // 1024 elements per wave
#define CHUNK      64                 // elements consumed per wmma iteration
#define ITERS      (WAVE_ELEMS / CHUNK)

#if __has_builtin(__builtin_amdgcn_wmma_f32_16x16x4_f32)
#define HAVE_WMMA_F32 1
#endif

typedef __attribute__((address_space(1))) v4i* g_v4i_p;   // global ptr to 16B vector
typedef __attribute__((address_space(3))) v4i* l_v4i_p;   // LDS ptr to 16B vector
typedef __attribute__((address_space(3))) void lds_void;

__device__ __forceinline__ void async_copy_b128(const float* gsrc, float* ldst) {
#if __has_builtin(__builtin_amdgcn_global_load_async_to_lds_b128)
    __builtin_amdgcn_global_load_async_to_lds_b128(
        (g_v4i_p)gsrc, (l_v4i_p)ldst, 0, 0);
#else
    uint32_t lofs = (uint32_t)(uintptr_t)(lds_void*)ldst;
    asm volatile("global_load_async_to_lds_b128 %0, %1, off"
                 :: "v"(lofs), "v"(gsrc) : "memory");
#endif
}

__device__ __forceinline__ void wait_async_all() {
#if __has_builtin(__builtin_amdgcn_s_wait_asynccnt)
    __builtin_amdgcn_s_wait_asynccnt(0);
#else
    asm volatile("s_wait_asynccnt 0" ::: "memory");
#endif
}

__device__ __forceinline__ float wave_allreduce_add(float v) {
    #pragma unroll
    for (int off = 16; off > 0; off >>= 1)
        v += __shfl_xor(v, off, 32);
    return v;
}

__global__ __launch_bounds__(BLOCK)
void ternary_rowquant_kernel(const float* __restrict__ w, float* __restrict__ out) {
    __shared__ float row[ROW];          // 32 KB: one full matrix row
    __shared__ float red0[WAVES];
    __shared__ float red1[WAVES];
    __shared__ float red2[WAVES];
    __shared__ float bc[2];             // [0]=delta, [1]=alpha

    const int tid  = threadIdx.x;
    const int wave = tid >> 5;
    const int lane = tid & 31;
    const long long rbase = (long long)blockIdx.x * ROW;
    const float* grow = w + rbase;

    // ---- Phase 0: async-stage the row into LDS (8 x b128 per thread) ----
    #pragma unroll
    for (int i = 0; i < ROW / (BLOCK * 4); ++i) {
        const int e = (i * BLOCK + tid) * 4;       // 16B-aligned, coalesced
        async_copy_b128(grow + e, &row[e]);
    }
    wait_async_all();
    __syncthreads();

    // ---- Phase 1: delta = clip(0.7 * mean(|w|), 0, 100) ----
    float delta;
    {
#ifdef HAVE_WMMA_F32
        v8f acc = {};
        const v2f bones = {1.0f, 1.0f};            // B = ones(4x16)
        const int m  = lane & 15;                  // A-matrix row
        const int k0 = (lane >> 4) << 1;           // K=0,1 (lanes 0-15) / K=2,3 (16-31)
        float s;
        {
            #pragma unroll 4
            for (int it = 0; it < ITERS; ++it) {
                const int base = wave * WAVE_ELEMS + it * CHUNK + m * 4 + k0;
                float2 d2 = *(const float2*)&row[base];   // ds_load_b64, no bank conflicts
                v2f a; a.x = fabsf(d2.x); a.y = fabsf(d2.y);
                acc = __builtin_amdgcn_wmma_f32_16x16x4_f32(
                          false, a, false, bones, (short)0, acc, false, false);
            }
            s = acc[0]+acc[1]+acc[2]+acc[3]+acc[4]+acc[5]+acc[6]+acc[7];
            s += __shfl_xor(s, 16, 32);            // join M=0..7 and M=8..15 halves
        }
#else
        float s = 0.0f;
        #pragma unroll
        for (int i = 0; i < WAVE_ELEMS / (32 * 4); ++i) {
            const int base = wave * WAVE_ELEMS + (i * 32 + lane) * 4;
            float4 v = *(const float4*)&row[base];
            s += fabsf(v.x) + fabsf(v.y) + fabsf(v.z) + fabsf(v.w);
        }
        s = wave_allreduce_add(s);
#endif
        if (lane == 0) red0[wave] = s;
        __syncthreads();
        if (tid == 0) {
            float t = 0.0f;
            #pragma unroll
            for (int i = 0; i < WAVES; ++i) t += red0[i];
            float d = 0.7f * t * (1.0f / (float)ROW);
            bc[0] = fminf(fmaxf(d, 0.0f), 100.0f);
        }
        __syncthreads();
        delta = bc[0];
    }

    // ---- Phase 2: alpha = clip(sum(|w| where |w|>delta) / count, 0, 100) ----
    float alpha;
    {
#ifdef HAVE_WMMA_F32
        v8f accA = {}, accC = {};
        const v2f bones = {1.0f, 1.0f};
        const int m  = lane & 15;
        const int k0 = (lane >> 4) << 1;
        float sA, sC;
        {
            #pragma unroll 4
            for (int it = 0; it < ITERS; ++it) {
                const int base = wave * WAVE_ELEMS + it * CHUNK + m * 4 + k0;
                float2 d2 = *(const float2*)&row[base];
                const float a0 = fabsf(d2.x), a1 = fabsf(d2.y);
                const bool m0 = a0 > delta, m1 = a1 > delta;
                v2f aa; aa.x = m0 ? a0 : 0.0f;   aa.y = m1 ? a1 : 0.0f;
                v2f ac; ac.x = m0 ? 1.0f : 0.0f; ac.y = m1 ? 1.0f : 0.0f;
                accA = __builtin_amdgcn_wmma_f32_16x16x4_f32(
                           false, aa, false, bones, (short)0, accA, false, false);
                accC = __builtin_amdgcn_wmma_f32_16x16x4_f32(
                           false, ac, false, bones, (short)0, accC, false, false);
            }
            sA = accA[0]+accA[1]+accA[2]+accA[3]+accA[4]+accA[5]+accA[6]+accA[7];
            sC = accC[0]+accC[1]+accC[2]+accC[3]+accC[4]+accC[5]+accC[6]+accC[7];
            sA += __shfl_xor(sA, 16, 32);
            sC += __shfl_xor(sC, 16, 32);
        }
#else
        float sA = 0.0f, sC = 0.0f;
        #pragma unroll
        for (int i = 0; i < WAVE_ELEMS / (32 * 4); ++i) {
            const int base = wave * WAVE_ELEMS + (i * 32 + lane) * 4;
            float4 v = *(const float4*)&row[base];
            float a0 = fabsf(v.x), a1 = fabsf(v.y), a2 = fabsf(v.z), a3 = fabsf(v.w);
            if (a0 > delta) { sA += a0; sC += 1.0f; }
            if (a1 > delta) { sA += a1; sC += 1.0f; }
            if (a2 > delta) { sA += a2; sC += 1.0f; }
            if (a3 > delta) { sA += a3; sC += 1.0f; }
        }
        sA = wave_allreduce_add(sA);
        sC = wave_allreduce_add(sC);
#endif
        if (lane == 0) { red1[wave] = sA; red2[wave] = sC; }
        __syncthreads();
        if (tid == 0) {
            float ta = 0.0f, tc = 0.0f;
            #pragma unroll
            for (int i = 0; i < WAVES; ++i) { ta += red1[i]; tc += red2[i]; }
            float al = ta / tc;
            bc[1] = fminf(fmaxf(al, 0.0f), 100.0f);
        }
        __syncthreads();
        alpha = bc[1];
    }

    // ---- Phase 3: out = alpha * ((w > delta) - (w < -delta)) ----
    const float nd = -delta;
    #pragma unroll
    for (int i = 0; i < ROW / (BLOCK * 4); ++i) {
        const int e = (i * BLOCK + tid) * 4;
        float4 v = *(const float4*)&row[e];
        float4 o;
        o.x = alpha * ((v.x > delta ? 1.0f : 0.0f) - (v.x < nd ? 1.0f : 0.0f));
        o.y = alpha * ((v.y > delta ? 1.0f : 0.0f) - (v.y < nd ? 1.0f : 0.0f));
        o.z = alpha * ((v.z > delta ? 1.0f : 0.0f) - (v.z < nd ? 1.0f : 0.0f));
        o.w = alpha * ((v.w > delta ? 1.0f : 0.0f) - (v.w < nd ? 1.0f : 0.0f));
        *(float4*)(out + rbase + e) = o;           // coalesced global_store_b128
    }
}

extern "C" void kernel_launch(void* const* d_in, const int* in_sizes, int n_in,
                              void* d_out, int out_size, void* d_ws, size_t ws_size,
                              hipStream_t stream) {
    (void)n_in; (void)out_size; (void)d_ws; (void)ws_size;
    const float* w = (const float*)d_in[0];
    float* out = (float*)d_out;
    const int rows = in_sizes[0] / ROW;            // 8192 rows
    hipLaunchKernelGGL(ternary_rowquant_kernel, dim3(rows), dim3(BLOCK), 0, stream,
                       w, out);
}